// ContainerGNN_38397007626339
// MI455X (gfx1250) — compile-verified
//
#include <hip/hip_runtime.h>
#include <cstdint>

typedef float v2f __attribute__((ext_vector_type(2)));
typedef float v8f __attribute__((ext_vector_type(8)));

#define BN_EPS  1e-5f
#define L2_EPS  1e-12f

__device__ __forceinline__ void atomAddF32(float* p, float v) {
    unsafeAtomicAdd(p, v);
}

// ---------------------------------------------------------------------------
__global__ void fill_f32_kernel(float* __restrict__ p, long n, float v) {
    long i = (long)blockIdx.x * blockDim.x + threadIdx.x;
    long stride = (long)gridDim.x * blockDim.x;
    for (; i < n; i += stride) p[i] = v;
}

__global__ void fill_i32_kernel(int* __restrict__ p, long n, int v) {
    long i = (long)blockIdx.x * blockDim.x + threadIdx.x;
    long stride = (long)gridDim.x * blockDim.x;
    for (; i < n; i += stride) p[i] = v;
}

// ---------------------------------------------------------------------------
// in-degree histogram over destinations (int atomics, once per call)
__global__ void count_kernel(const long long* __restrict__ dst, int* __restrict__ cnt, int nE) {
    int i = blockIdx.x * blockDim.x + threadIdx.x;
    int stride = gridDim.x * blockDim.x;
    for (; i < nE; i += stride)
        atomicAdd(&cnt[(int)dst[i]], 1);
}

// dinv = rsqrt(cnt + 1)  (self loop included in degree)
__global__ void dinv_kernel(const int* __restrict__ cnt, float* __restrict__ dinv, int Nn) {
    int i = blockIdx.x * blockDim.x + threadIdx.x;
    if (i < Nn) dinv[i] = rsqrtf((float)cnt[i] + 1.0f);
}

// single-block exclusive scan: ptr[0]=0, ptr[i+1]=incl(i), cursor[i]=excl(i)
__global__ void scan_kernel(const int* __restrict__ cnt, int* __restrict__ ptr,
                            int* __restrict__ cursor, int Nn) {
    __shared__ int smem[1024];
    __shared__ int carry;
    if (threadIdx.x == 0) { carry = 0; ptr[0] = 0; }
    __syncthreads();
    for (int base = 0; base < Nn; base += 1024) {
        int i = base + (int)threadIdx.x;
        int v = (i < Nn) ? cnt[i] : 0;
        smem[threadIdx.x] = v;
        __syncthreads();
        for (int off = 1; off < 1024; off <<= 1) {
            int t = (threadIdx.x >= (unsigned)off) ? smem[threadIdx.x - off] : 0;
            __syncthreads();
            smem[threadIdx.x] += t;
            __syncthreads();
        }
        int incl = smem[threadIdx.x] + carry;
        if (i < Nn) { ptr[i + 1] = incl; cursor[i] = incl - v; }
        __syncthreads();
        if (threadIdx.x == 1023) carry = incl;
        __syncthreads();
    }
}

// bucket edges by destination: csr_src[pos] = src
__global__ void scatter_kernel(const long long* __restrict__ src, const long long* __restrict__ dst,
                               int* __restrict__ cursor, int* __restrict__ csr_src, int nE) {
    int i = blockIdx.x * blockDim.x + threadIdx.x;
    int stride = gridDim.x * blockDim.x;
    for (; i < nE; i += stride) {
        int d = (int)dst[i];
        int pos = atomicAdd(&cursor[d], 1);
        csr_src[pos] = (int)src[i];
    }
}

// ---------------------------------------------------------------------------
// C[M,N] = act(A[M,K] @ W[K,N] + bias)   via V_WMMA_F32_16X16X4_F32
// one wave computes a 16x16 tile; M,N multiples of 16; K multiple of 4.
template<int RELU>
__global__ void gemm_wmma_f32(const float* __restrict__ A, const float* __restrict__ W,
                              const float* __restrict__ bias, float* __restrict__ C,
                              int M, int K, int N) {
    int wid  = (int)((blockIdx.x * blockDim.x + threadIdx.x) >> 5);
    int lane = threadIdx.x & 31;
    int nT = N >> 4;
    int mT = M >> 4;
    if (wid >= mT * nT) return;               // wave-uniform: EXEC stays all-ones
    int mt = wid / nT, nt = wid - mt * nT;
    int mbase = mt << 4, nbase = nt << 4;
    int half = lane >> 4;                     // 0: lanes 0-15, 1: lanes 16-31
    int l16  = lane & 15;
    int arow = mbase + l16;                   // A row for this lane (M index)
    int ncol = nbase + l16;                   // B/C column for this lane (N index)

    v8f acc = {0.f,0.f,0.f,0.f,0.f,0.f,0.f,0.f};
    for (int k = 0; k < K; k += 4) {
        int ka = k + half * 2;                // lanes 0-15: K=k,k+1; 16-31: K=k+2,k+3
        v2f a, b;
        a.x = A[(long)arow * K + ka];
        a.y = A[(long)arow * K + ka + 1];
        b.x = W[(long)ka * N + ncol];
        b.y = W[(long)(ka + 1) * N + ncol];
        acc = __builtin_amdgcn_wmma_f32_16x16x4_f32(
            /*neg_a=*/false, a, /*neg_b=*/false, b,
            /*c_mod=*/(short)0, acc, /*reuse_a=*/false, /*reuse_b=*/false);
    }
    float bv = bias ? bias[ncol] : 0.0f;
    int rbase = mbase + half * 8;             // D layout: VGPR i -> M = half*8 + i
#pragma unroll
    for (int i = 0; i < 8; ++i) {
        float v = acc[i] + bv;
        if (RELU) v = fmaxf(v, 0.0f);
        C[(long)(rbase + i) * N + ncol] = v;
    }
}

// ---------------------------------------------------------------------------
// CSR gather aggregation, fused with self-loop and bias:
//   out[n] = sum_{s in nbr(n)} T[s]*dinv[s]*dinv[n] + T[n]*dinv[n]^2 + bias
// one wave per node; lane owns CPL consecutive columns (D = 32*CPL).
// neighbor index / dinv loads are wave-uniform; row reads are fully coalesced.
template<int CPL>
__global__ void gcn_gather_kernel(const int* __restrict__ ptr, const int* __restrict__ csr_src,
                                  const float* __restrict__ dinv, const float* __restrict__ T,
                                  const float* __restrict__ bias, float* __restrict__ out,
                                  int Nn) {
    const int D = 32 * CPL;
    int n    = (int)((blockIdx.x * blockDim.x + threadIdx.x) >> 5);
    int lane = threadIdx.x & 31;
    if (n >= Nn) return;                      // wave-uniform
    int c0 = lane * CPL;
    float dn = dinv[n];
    float acc[CPL];
#pragma unroll
    for (int j = 0; j < CPL; ++j) acc[j] = 0.0f;

    int beg = ptr[n], end = ptr[n + 1];
    for (int i = beg; i < end; ++i) {
        int s = csr_src[i];                   // wave-uniform
        float c = dinv[s] * dn;
        const float* hs = T + (long)s * D + c0;
#pragma unroll
        for (int j = 0; j < CPL; ++j) acc[j] += hs[j] * c;
    }
    // self loop + bias
    const float* hn = T + (long)n * D + c0;
    float d2 = dn * dn;
    float* on = out + (long)n * D + c0;
#pragma unroll
    for (int j = 0; j < CPL; ++j)
        on[j] = acc[j] + hn[j] * d2 + bias[c0 + j];
}

// ---------------------------------------------------------------------------
// column sums and sum-of-squares (stats[0..D-1]=sum, stats[D..2D-1]=sumsq)
__global__ void colstats_kernel(const float* __restrict__ X, float* __restrict__ stats,
                                int Nn, int D) {
    int col = threadIdx.x % D;                 // D divides blockDim (256)
    int rpb = blockDim.x / D;
    int r   = blockIdx.x * rpb + threadIdx.x / D;
    int stride = gridDim.x * rpb;
    float s = 0.f, sq = 0.f;
    for (; r < Nn; r += stride) {
        float v = X[(long)r * D + col];
        s += v; sq += v * v;
    }
    atomAddF32(&stats[col], s);
    atomAddF32(&stats[D + col], sq);
}

// fold BN into per-column affine: scale = g*rsqrt(var+eps); shift = beta - mu*scale
__global__ void bn_finalize_kernel(const float* __restrict__ stats,
                                   const float* __restrict__ g, const float* __restrict__ be,
                                   float* __restrict__ scale, float* __restrict__ shift,
                                   int D, float invN) {
    int c = blockIdx.x * blockDim.x + threadIdx.x;
    if (c >= D) return;
    float mu  = stats[c] * invN;
    float var = stats[D + c] * invN - mu * mu;
    float sc  = g[c] * rsqrtf(var + BN_EPS);
    scale[c] = sc;
    shift[c] = be[c] - mu * sc;
}

template<int RELU>
__global__ void affine_kernel(float* __restrict__ X, const float* __restrict__ scale,
                              const float* __restrict__ shift, int Nn, int D) {
    long i = (long)blockIdx.x * blockDim.x + threadIdx.x;
    long total = (long)Nn * D;
    if (i >= total) return;
    int col = (int)(i % D);
    float v = X[i] * scale[col] + shift[col];
    if (RELU) v = fmaxf(v, 0.0f);
    X[i] = v;
}

// ---------------------------------------------------------------------------
// row-wise L2 normalize, D = 32 (one wave per row, shuffle reduction)
__global__ void l2norm_kernel(const float* __restrict__ X, float* __restrict__ out, int Nn) {
    int wid  = (int)((blockIdx.x * blockDim.x + threadIdx.x) >> 5);
    int lane = threadIdx.x & 31;
    if (wid >= Nn) return;                     // wave-uniform
    float v  = X[(long)wid * 32 + lane];
    float sq = v * v;
#pragma unroll
    for (int off = 16; off > 0; off >>= 1)
        sq += __shfl_xor(sq, off, 32);
    float nrm = sqrtf(sq);
    out[(long)wid * 32 + lane] = v / fmaxf(nrm, L2_EPS);
}

// ---------------------------------------------------------------------------
extern "C" void kernel_launch(void* const* d_in, const int* in_sizes, int n_in,
                              void* d_out, int out_size, void* d_ws, size_t ws_size,
                              hipStream_t stream) {
    const float*     x     = (const float*)d_in[0];
    const long long* ei    = (const long long*)d_in[1];
    const float*     ne_w1 = (const float*)d_in[2];
    const float*     ne_b1 = (const float*)d_in[3];
    const float*     ne_g  = (const float*)d_in[4];
    const float*     ne_be = (const float*)d_in[5];
    const float*     ne_w2 = (const float*)d_in[6];
    const float*     ne_b2 = (const float*)d_in[7];
    const float*     c1_w  = (const float*)d_in[8];
    const float*     c1_b  = (const float*)d_in[9];
    const float*     g1    = (const float*)d_in[10];
    const float*     be1   = (const float*)d_in[11];
    const float*     c2_w  = (const float*)d_in[12];
    const float*     c2_b  = (const float*)d_in[13];
    const float*     g2    = (const float*)d_in[14];
    const float*     be2   = (const float*)d_in[15];
    const float*     c3_w  = (const float*)d_in[16];
    const float*     c3_b  = (const float*)d_in[17];
    const float*     g3    = (const float*)d_in[18];
    const float*     be3   = (const float*)d_in[19];

    const int Nn = in_sizes[0] / 32;           // 50000
    const int nE = in_sizes[1] / 2;            // 1,600,000
    const long long* e_src = ei;
    const long long* e_dst = ei + nE;
    const int NnP = (Nn + 63) & ~63;

    // workspace layout
    float* ws    = (float*)d_ws;
    float* dinv  = ws;                         // [NnP]
    float* T     = dinv + NnP;                 // [Nn*64] gemm output
    float* H     = T    + (long)Nn * 64;       // [Nn*64] current features
    float* stats = H    + (long)Nn * 64;       // [128]
    float* scale = stats + 128;                // [64]
    float* shift = scale + 64;                 // [64]
    int*   cnt     = (int*)(shift + 64);       // [NnP]
    int*   csr_ptr = cnt + NnP;                // [Nn+1]
    int*   cursor  = csr_ptr + NnP;            // [Nn]
    int*   csr_src = cursor + NnP;             // [nE]

    const float invN = 1.0f / (float)Nn;
    const int TB = 256;
    auto blocks1d = [&](long n) { return (int)((n + TB - 1) / TB); };
    auto gemm_blocks = [&](int M, int N) { return ((M >> 4) * (N >> 4) + 3) / 4; };

    // ---- CSR build: degree count -> dinv, scan -> scatter -------------------
    fill_i32_kernel<<<blocks1d(Nn), TB, 0, stream>>>(cnt, Nn, 0);
    count_kernel<<<2048, TB, 0, stream>>>(e_dst, cnt, nE);
    dinv_kernel<<<blocks1d(Nn), TB, 0, stream>>>(cnt, dinv, Nn);
    scan_kernel<<<1, 1024, 0, stream>>>(cnt, csr_ptr, cursor, Nn);
    scatter_kernel<<<2048, TB, 0, stream>>>(e_src, e_dst, cursor, csr_src, nE);

    // ---- node encoder -------------------------------------------------------
    // T = relu(x @ ne_w1 + ne_b1); BN affine in place; H = T @ ne_w2 + ne_b2
    gemm_wmma_f32<1><<<gemm_blocks(Nn, 64), 128, 0, stream>>>(x, ne_w1, ne_b1, T, Nn, 32, 64);
    fill_f32_kernel<<<1, 128, 0, stream>>>(stats, 128, 0.0f);
    colstats_kernel<<<512, TB, 0, stream>>>(T, stats, Nn, 64);
    bn_finalize_kernel<<<1, 64, 0, stream>>>(stats, ne_g, ne_be, scale, shift, 64, invN);
    affine_kernel<0><<<blocks1d((long)Nn * 64), TB, 0, stream>>>(T, scale, shift, Nn, 64);
    gemm_wmma_f32<0><<<gemm_blocks(Nn, 64), 128, 0, stream>>>(T, ne_w2, ne_b2, H, Nn, 64, 64);

    // ---- GCN conv layers (gather aggregation, no float atomics) -------------
    struct Layer { const float *W, *b, *g, *be; int Dout; int relu; };
    Layer layers[3] = {
        { c1_w, c1_b, g1, be1, 64, 1 },
        { c2_w, c2_b, g2, be2, 64, 1 },
        { c3_w, c3_b, g3, be3, 32, 0 },
    };
    int Din = 64;
    const int gatherBlocks = (Nn + 7) / 8;     // one wave per node, 8 waves/block
    for (int l = 0; l < 3; ++l) {
        const Layer& L = layers[l];
        // T = H @ W
        gemm_wmma_f32<0><<<gemm_blocks(Nn, L.Dout), 128, 0, stream>>>(
            H, L.W, nullptr, T, Nn, Din, L.Dout);
        // H = gather(T) + self + bias
        if (L.Dout == 64)
            gcn_gather_kernel<2><<<gatherBlocks, TB, 0, stream>>>(
                csr_ptr, csr_src, dinv, T, L.b, H, Nn);
        else
            gcn_gather_kernel<1><<<gatherBlocks, TB, 0, stream>>>(
                csr_ptr, csr_src, dinv, T, L.b, H, Nn);
        // BN -> affine (+relu) in place
        fill_f32_kernel<<<1, 128, 0, stream>>>(stats, 128, 0.0f);
        colstats_kernel<<<512, TB, 0, stream>>>(H, stats, Nn, L.Dout);
        bn_finalize_kernel<<<1, 64, 0, stream>>>(stats, L.g, L.be, scale, shift, L.Dout, invN);
        if (L.relu)
            affine_kernel<1><<<blocks1d((long)Nn * L.Dout), TB, 0, stream>>>(H, scale, shift, Nn, L.Dout);
        else
            affine_kernel<0><<<blocks1d((long)Nn * L.Dout), TB, 0, stream>>>(H, scale, shift, Nn, L.Dout);
        Din = L.Dout;
    }

    // ---- final row-wise L2 normalize (D = 32) -------------------------------
    l2norm_kernel<<<(Nn + 7) / 8, TB, 0, stream>>>(H, (float*)d_out, Nn);
}